// EuclCodebook_75488345194613
// MI455X (gfx1250) — compile-verified
//
#include <hip/hip_runtime.h>
#include <hip/hip_bf16.h>

#define DIMC 256
#define KCODES 2048
#define BN 65536               // 32*2048 tokens
#define TOK_PER_WAVE 16
#define WAVES_PER_BLOCK 8
#define TOK_PER_BLOCK (TOK_PER_WAVE * WAVES_PER_BLOCK)   // 128
#define ROWF 260               // staged row stride in floats (256 + 4 pad -> bank-conflict-free)
#define TILEF (16 * ROWF)      // floats per staged emb tile buffer (16.25 KB)

typedef __attribute__((ext_vector_type(2))) float v2f;
typedef __attribute__((ext_vector_type(8))) float v8f;
typedef __attribute__((ext_vector_type(4))) unsigned int u32x4;
typedef __attribute__((ext_vector_type(8))) int i32x8;
typedef __attribute__((ext_vector_type(4))) int i32x4;

#if __has_builtin(__builtin_amdgcn_tensor_load_to_lds)
#define USE_TDM 1
#endif

#ifdef USE_TDM
// ---------------------------------------------------------------------------
// One TDM DMA: stage a 16-row x 256-float emb tile into LDS, inserting a
// 16-byte pad after every row (pad_interval = 256 dwords, pad_amount = 4
// dwords) so the staged row stride is ROWF floats.
// D# group0: count=1, lds_addr, 57-bit global_addr, type=2 ("image").
// D# group1: data_size=4B, tensor 256x16, tile 256x16, dim0 stride 256.
// ---------------------------------------------------------------------------
__device__ __forceinline__ void tdm_stage_tile(unsigned long long gaddr,
                                               unsigned int ldsaddr) {
  u32x4 g0 = {1u,                                           // count=1, no gather
              ldsaddr,                                      // lds_addr [63:32]
              (unsigned int)(gaddr & 0xFFFFFFFFu),          // global_addr lo
              (unsigned int)((gaddr >> 32) & 0x01FFFFFFu)   // global_addr hi (57b)
                  | 0x80000000u};                           // type = 2 -> bit127
  i32x8 g1 = {(int)((2u << 16)       // data_size = 4 bytes
                    | (1u << 20)     // pad_enable
                    | (7u << 22)     // pad_interval: 256 dwords
                    | (3u << 25)),   // pad_amount: 4 dwords (16 B)
              (int)(256u << 16),     // tensor_dim0 = 256 (bits 79:48, low part)
              (int)(16u << 16),      // tensor_dim1 = 16  (bits 111:80)
              (int)(256u << 16),     // tile_dim0 = 256   (bits 127:112)
              16,                    // tile_dim1 = 16    (bits 143:128)
              256,                   // tensor_dim0_stride = 256 (bits 207:160)
              0, 0};
  i32x4 gz = {0, 0, 0, 0};
#if __clang_major__ >= 23
  i32x8 gz8 = {0, 0, 0, 0, 0, 0, 0, 0};
  __builtin_amdgcn_tensor_load_to_lds(g0, g1, gz, gz, gz8, 0);
#else
  __builtin_amdgcn_tensor_load_to_lds(g0, g1, gz, gz, 0);
#endif
}
#endif

// ---------------------------------------------------------------------------
// Kernel 1: codebook row norms ||e_k||^2  (2048 rows x 256) — one wave per row
// ---------------------------------------------------------------------------
__global__ __launch_bounds__(256) void EuclCodebook_norm_kernel(
    const float* __restrict__ emb, float* __restrict__ norms) {
  const int wave = threadIdx.x >> 5;
  const int lane = threadIdx.x & 31;
  const int row  = blockIdx.x * WAVES_PER_BLOCK + wave;
  const float4* r = (const float4*)(emb + (long)row * DIMC);
  float s = 0.0f;
#pragma unroll
  for (int i = 0; i < 2; ++i) {
    float4 v = r[lane + i * 32];
    s += v.x * v.x + v.y * v.y + v.z * v.z + v.w * v.w;
  }
#pragma unroll
  for (int off = 16; off > 0; off >>= 1) s += __shfl_xor(s, off, 32);
  if (lane == 0) norms[row] = s;
}

// ---------------------------------------------------------------------------
// Kernel 2: WMMA distance GEMM + per-token argmin.
// Transposed tile: A = emb codes (16x4) staged in LDS by the TDM,
// B = (-2*z) tokens (4x16) resident in VGPRs, C preloaded with ||e||^2 so
// acc = ||e||^2 - 2 z.e (argmin-equivalent score). Each lane holds 8 scores
// for ONE token -> pure in-lane argmin; one shfl_xor(16) merge at the end.
// emb tiles are double-buffered: wave 0 issues the next tile's DMA, waits
// TENSORcnt <= 1, and workgroup barriers publish/recycle buffers.
// ---------------------------------------------------------------------------
__global__ __launch_bounds__(256) void EuclCodebook_argmin_kernel(
    const float* __restrict__ z, const float* __restrict__ emb,
    const float* __restrict__ norms, int* __restrict__ idx_out,
    float* __restrict__ idx_out_f) {
  __shared__ float snorm[KCODES];
#ifdef USE_TDM
  __shared__ float abuf[2][TILEF];
#else
  __shared__ float abuf[1][TILEF];
#endif

  for (int i = threadIdx.x; i < KCODES; i += 256) snorm[i] = norms[i];

  const int wave  = threadIdx.x >> 5;
  const int lane  = threadIdx.x & 31;
  const int half  = lane >> 4;       // 0: K pair {0,1}/codes r ; 1: {2,3}/codes r+8
  const int lanem = lane & 15;
  const int tok = blockIdx.x * TOK_PER_BLOCK + wave * TOK_PER_WAVE + lanem;

  // B fragments: this lane's token row of z, scaled by -2, resident in VGPRs.
  // B layout (4x16): V0 = B[2*half][n], V1 = B[2*half+1][n], n = lanem.
  v2f bfrag[64];
  const float* zrow = z + (long)tok * DIMC + half * 2;
#pragma unroll
  for (int d = 0; d < 64; ++d) {
    v2f t = *(const v2f*)(zrow + d * 4);
    bfrag[d].x = -2.0f * t.x;
    bfrag[d].y = -2.0f * t.y;
  }

  float bestS = 3.0e38f;
  int   bestI = 0;

#ifdef USE_TDM
  const unsigned long long embq = (unsigned long long)(uintptr_t)emb;
  const unsigned int lds0 = (unsigned int)(uintptr_t)(&abuf[0][0]);
  const unsigned int lds1 = (unsigned int)(uintptr_t)(&abuf[1][0]);
  if (wave == 0) tdm_stage_tile(embq, lds0);           // prologue: tile 0 -> buf0
#endif

  for (int t = 0; t < KCODES / 16; ++t) {
    const int c0 = t * 16;

#ifdef USE_TDM
    if (wave == 0) {
      if (t + 1 < KCODES / 16) {
        // stage tile t+1 into buf[(t+1)&1]; that buffer was last read at
        // iteration t-1 and released by the barrier at the end of it.
        tdm_stage_tile(embq + (unsigned long long)(c0 + 16) * (DIMC * 4ull),
                       (t & 1) ? lds0 : lds1);
        __builtin_amdgcn_s_wait_tensorcnt(1);          // tile t DMA complete
      } else {
        __builtin_amdgcn_s_wait_tensorcnt(0);
      }
    }
    __syncthreads();                                   // publish tile t
    const float* ap = &abuf[t & 1][lanem * ROWF + half * 2];
#else
    __syncthreads();
    {  // synchronous cooperative staging fallback (single buffer)
      const int m = threadIdx.x >> 4;
      const int ch = threadIdx.x & 15;
      const float4* src = (const float4*)(emb + (long)(c0 + m) * DIMC + ch * 16);
      float4* dst = (float4*)(&abuf[0][m * ROWF + ch * 16]);
#pragma unroll
      for (int j = 0; j < 4; ++j) dst[j] = src[j];
    }
    __syncthreads();
    const float* ap = &abuf[0][lanem * ROWF + half * 2];
#endif

    // C init: component r <-> code c0 + r + 8*half (transposed C layout)
    v8f acc;
#pragma unroll
    for (int r = 0; r < 8; ++r) acc[r] = snorm[c0 + r + half * 8];

    // A from LDS: lane (m=lanem, half) reads word m*ROWF + 4d + 2*half;
    // ROWF = 260 ≡ 4 (mod 64) makes the 64 requested words hit 64 distinct banks.
#pragma unroll
    for (int d = 0; d < 64; ++d) {
      v2f a = *(const v2f*)(ap + d * 4);
      acc = __builtin_amdgcn_wmma_f32_16x16x4_f32(
          /*neg_a=*/false, a, /*neg_b=*/false, bfrag[d],
          /*c_mod=*/(short)0, acc, /*reuse_a=*/false, /*reuse_b=*/false);
    }

    // Candidates arrive in increasing code index, so strict '<' preserves
    // jnp.argmin's first-min tie rule within a lane.
#pragma unroll
    for (int r = 0; r < 8; ++r) {
      float s = acc[r];
      int   ci = c0 + r + half * 8;
      if (s < bestS) { bestS = s; bestI = ci; }
    }

#ifdef USE_TDM
    __syncthreads();   // all waves done reading buf[t&1] before it is re-staged
#endif
  }

  // Merge the two half-wave code sets for each token (lexicographic tie-break).
  float oS = __shfl_xor(bestS, 16, 32);
  int   oI = __shfl_xor(bestI, 16, 32);
  if (oS < bestS || (oS == bestS && oI < bestI)) { bestS = oS; bestI = oI; }

  if (half == 0) {
    idx_out[tok]   = bestI;
    idx_out_f[tok] = (float)bestI;
  }
}

// ---------------------------------------------------------------------------
// Kernel 3: gather z_q = emb[idx] (== z_q_st in fwd), per-token SSE partial.
// One 64-thread block per token, float4 per thread.
// ---------------------------------------------------------------------------
__global__ __launch_bounds__(64) void EuclCodebook_gather_kernel(
    const float* __restrict__ z, const float* __restrict__ emb,
    const int* __restrict__ idx, float* __restrict__ zq_out,
    float* __restrict__ partials) {
  __shared__ float wsum[2];
  const long t   = blockIdx.x;
  const int  tid = threadIdx.x;
  const int  code = idx[t];

  float4 e  = ((const float4*)(emb + (long)code * DIMC))[tid];
  float4 zz = ((const float4*)(z + t * DIMC))[tid];
  ((float4*)(zq_out + t * DIMC))[tid] = e;   // z + sg(z_q - z) == z_q in fwd

  float dx = e.x - zz.x, dy = e.y - zz.y, dz = e.z - zz.z, dw = e.w - zz.w;
  float s = dx * dx + dy * dy + dz * dz + dw * dw;
#pragma unroll
  for (int off = 16; off > 0; off >>= 1) s += __shfl_xor(s, off, 32);
  if ((tid & 31) == 0) wsum[tid >> 5] = s;
  __syncthreads();
  if (tid == 0) partials[t] = wsum[0] + wsum[1];
}

// ---------------------------------------------------------------------------
// Kernel 4: deterministic final reduction -> loss = 2 * SSE / (B*N*D)
// ---------------------------------------------------------------------------
__global__ __launch_bounds__(256) void EuclCodebook_loss_kernel(
    const float* __restrict__ partials, float* __restrict__ out_loss) {
  __shared__ float sred[256];
  float s = 0.0f;
  for (int i = threadIdx.x; i < BN; i += 256) s += partials[i];
  sred[threadIdx.x] = s;
  __syncthreads();
  for (int stride = 128; stride > 0; stride >>= 1) {
    if (threadIdx.x < stride) sred[threadIdx.x] += sred[threadIdx.x + stride];
    __syncthreads();
  }
  if (threadIdx.x == 0)
    out_loss[0] = 2.0f * sred[0] / (float)(BN * DIMC);
}

extern "C" void kernel_launch(void* const* d_in, const int* in_sizes, int n_in,
                              void* d_out, int out_size, void* d_ws, size_t ws_size,
                              hipStream_t stream) {
  const float* z   = (const float*)d_in[0];   // [32,2048,256] f32
  const float* emb = (const float*)d_in[1];   // [2048,256]    f32

  float* out   = (float*)d_out;
  float* zq    = out;                          // 16,777,216 floats
  float* idxf  = out + (long)BN * DIMC;        // 65,536 floats (indices as f32)
  float* lossp = idxf + BN;                    // 1 float

  char*  ws       = (char*)d_ws;
  float* norms    = (float*)ws;                         // 2048 f32   (8 KB)
  int*   idxbuf   = (int*)(ws + 8192);                  // 65536 i32  (256 KB)
  float* partials = (float*)(ws + 8192 + BN * 4);       // 65536 f32  (256 KB)

  EuclCodebook_norm_kernel<<<KCODES / WAVES_PER_BLOCK, 256, 0, stream>>>(emb, norms);
  EuclCodebook_argmin_kernel<<<BN / TOK_PER_BLOCK, 256, 0, stream>>>(
      z, emb, norms, idxbuf, idxf);
  EuclCodebook_gather_kernel<<<BN, 64, 0, stream>>>(z, emb, idxbuf, zq, partials);
  EuclCodebook_loss_kernel<<<1, 256, 0, stream>>>(partials, lossp);
}